// LossAF_39994735460664
// MI455X (gfx1250) — compile-verified
//
#include <hip/hip_runtime.h>
#include <float.h>
#include <math.h>

// ---------------- problem constants (match reference) ----------------
#define NCLS   80
#define T_ANCH 8400          // 6400 + 1600 + 400
#define BATCH  32
#define NGT    20
#define TOPK   10
#define NCAND  (NGT*TOPK)    // 200
#define NCOST  (NCAND*NGT)   // 4000
#define EPSF   1e-7f
#define INV_PI2 0.405284734569351f   // 4/pi^2

#define ABLK   128           // threads (= anchors) per block in kernel A

typedef float v2f __attribute__((ext_vector_type(2)));
typedef float v8f __attribute__((ext_vector_type(8)));
typedef int   v4i __attribute__((vector_size(16)));            // matches builtin param type
typedef __attribute__((address_space(1))) v4i* gv4i_p;         // global (AS1) v4i*
typedef __attribute__((address_space(3))) v4i* lv4i_p;         // LDS (AS3) v4i*

#if __has_builtin(__builtin_amdgcn_global_load_async_to_lds_b128) && \
    __has_builtin(__builtin_amdgcn_s_wait_asynccnt)
#define HAVE_ASYNC_LDS 1
#else
#define HAVE_ASYNC_LDS 0
#endif

// ---------------- math helpers ----------------
__device__ __forceinline__ float sigm(float x){ return 1.0f/(1.0f+expf(-x)); }
__device__ __forceinline__ float softplusf(float x){ return fmaxf(x,0.0f)+log1pf(expf(-fabsf(x))); }
// focal_map with target 0: 0.75 * softplus(x) * sigmoid(x)^2
__device__ __forceinline__ float focal0f(float x){ float s=sigm(x); return 0.75f*softplusf(x)*s*s; }
// focal_map with target 1: 0.25 * softplus(-x) * (1-sigmoid(x))^2
__device__ __forceinline__ float focal1f(float x){ float s=sigm(x); float q=1.0f-s; return 0.25f*softplusf(-x)*q*q; }

// flat anchor t (0..8399) -> level geometry
__device__ __forceinline__ void anchor_meta(int t,int& S,float& stride,int& cell){
  if (t < 6400){ S=80; stride= 8.f; cell=t; }
  else if (t < 8000){ S=40; stride=16.f; cell=t-6400; }
  else { S=20; stride=32.f; cell=t-8000; }
}

__device__ __forceinline__ const float* anchor_gptr(const float* p0,const float* p1,const float* p2,
                                                    int b,int t){
  if (t < 6400)      return p0 + ((size_t)b*6400 + t)*84;
  else if (t < 8000) return p1 + ((size_t)b*1600 + (t-6400))*84;
  else               return p2 + ((size_t)b*400  + (t-8000))*84;
}

__device__ __forceinline__ float logit_at(const float* p0,const float* p1,const float* p2,
                                          int b,int t,int c){
  return anchor_gptr(p0,p1,p2,b,t)[4+c];
}

__device__ float bbox_ciou_f(float px1,float py1,float px2,float py2,
                             float tx1,float ty1,float tx2,float ty2){
  float pw=fmaxf(px2-px1,EPSF), ph=fmaxf(py2-py1,EPSF);
  float tw=fmaxf(tx2-tx1,EPSF), th=fmaxf(ty2-ty1,EPSF);
  float iw=fmaxf(fminf(px2,tx2)-fmaxf(px1,tx1),0.f);
  float ih=fmaxf(fminf(py2,ty2)-fmaxf(py1,ty1),0.f);
  float inter=iw*ih;
  float uni=pw*ph+tw*th-inter+EPSF;
  float iou=inter/uni;
  float dx=(px1+px2)*0.5f-(tx1+tx2)*0.5f;
  float dy=(py1+py2)*0.5f-(ty1+ty2)*0.5f;
  float cd=dx*dx+dy*dy;
  float cw=fmaxf(px2,tx2)-fminf(px1,tx1);
  float ch=fmaxf(py2,ty2)-fminf(py1,ty1);
  float c2=cw*cw+ch*ch+EPSF;
  float dv=atanf(tw/th)-atanf(pw/ph);
  float v=INV_PI2*dv*dv;
  float alpha=v/(v-iou+1.0f+EPSF);
  return iou - cd/c2 - alpha*v;
}

// =====================================================================
// Kernel A: decode boxes + fused focal(x, t=0) partial sums.
// Each wave stages its 32 anchors (32*336B) into LDS with coalesced
// per-lane GLOBAL_LOAD_ASYNC_TO_LDS_B128 (ASYNCcnt-tracked), waits
// s_wait_asynccnt 0, then decodes out of LDS. Block partials reduced
// with WMMA f32 16x16x4 (B = all-ones, layout-proof on the B side).
// =====================================================================
__global__ __launch_bounds__(ABLK)
void decode_focal_kernel(const float* __restrict__ p0, const float* __restrict__ p1,
                         const float* __restrict__ p2,
                         float4* __restrict__ boxes, float* __restrict__ blockSums){
  __shared__ float sbuf[ABLK*84];    // 43008 B staging buffer
  __shared__ float partial[ABLK];

  const int tid  = threadIdx.x;
  const int gid  = blockIdx.x*ABLK + tid;     // grid covers BATCH*T_ANCH exactly
  const int b    = gid / T_ANCH;
  const int t    = gid - b*T_ANCH;
  const int wave = tid >> 5;
  const int lane = tid & 31;
  float* lbuf = sbuf + wave*(32*84);

#if HAVE_ASYNC_LDS
  {
    const int waveBase = blockIdx.x*ABLK + wave*32;   // first flat (b,t) of this wave
    #pragma unroll
    for (int i=0;i<21;i++){
      int j = i*32 + lane;            // 0..671 : 16B chunks of the wave's region
      int a = j/21, c = j - 21*a;     // anchor-in-wave, chunk-in-anchor
      int g2 = waveBase + a;
      int b2 = g2 / T_ANCH, t2 = g2 - b2*T_ANCH;
      const float* apg = anchor_gptr(p0,p1,p2,b2,t2);
      __builtin_amdgcn_global_load_async_to_lds_b128(
          (gv4i_p)(apg + c*4),
          (lv4i_p)(lbuf + a*84 + c*4),
          0, 0);
    }
    __builtin_amdgcn_s_wait_asynccnt(0);
  }
  const float* ap = lbuf + lane*84;   // this thread's anchor, now in LDS
#else
  const float* ap = anchor_gptr(p0,p1,p2,b,t);
  __builtin_prefetch(ap, 0, 1);
  __builtin_prefetch(ap + 64, 0, 1);
#endif

  int S, cell; float stride;
  anchor_meta(t, S, stride, cell);
  int gy = cell / S, gx = cell - gy*S;
  float tx=ap[0], ty=ap[1], tw=ap[2], th=ap[3];
  float px=(sigm(tx)*2.0f-0.5f+(float)gx)*stride;
  float py=(sigm(ty)*2.0f-0.5f+(float)gy)*stride;
  float pw=softplusf(tw)*stride;
  float ph=softplusf(th)*stride;
  boxes[(size_t)b*T_ANCH + t] = make_float4(px,py,pw,ph);

  float acc=0.f;
  #pragma unroll 8
  for (int c=0;c<NCLS;c++) acc += focal0f(ap[4+c]);

  partial[tid]=acc;
  __syncthreads();

#if __has_builtin(__builtin_amdgcn_wmma_f32_16x16x4_f32)
  // Wave 0 reduces ABLK partials: (ABLK/64) WMMAs with B = ones.
  if (tid < 32){
    int m = lane & 15;
    int kbase = (lane < 16) ? 0 : 2;            // ISA 16x4 f32 A layout
    v2f ones; ones[0]=1.0f; ones[1]=1.0f;
    v8f c = {};
    #pragma unroll
    for (int i=0;i<ABLK/64;i++){
      v2f a;
      a[0]=partial[i*64 + m*4 + kbase];
      a[1]=partial[i*64 + m*4 + kbase + 1];
      c = __builtin_amdgcn_wmma_f32_16x16x4_f32(false, a, false, ones,
                                                (short)0, c, false, false);
    }
    float s = c[0]+c[1]+c[2]+c[3]+c[4]+c[5]+c[6]+c[7];
    float tot = __shfl(s, 0, 32) + __shfl(s, 16, 32);  // D column 0: lane0 m0..7, lane16 m8..15
    if (lane==0) blockSums[blockIdx.x]=tot;
  }
#else
  for (int off=ABLK/2; off>0; off>>=1){
    if (tid < off) partial[tid]+=partial[tid+off];
    __syncthreads();
  }
  if (tid==0) blockSums[blockIdx.x]=partial[0];
#endif
}

// =====================================================================
// Kernel B: per-image matching (one block per image, 256 threads).
// dist -> top-10/GT (iterative block-argmin over LDS-resident dist),
// 200x20 cost matrix in LDS, greedy assignment by (cost, flat-idx)
// lexicographic min (== stable-sorted scan), then CIoU + focal correction.
// =====================================================================
__global__ __launch_bounds__(256)
void match_kernel(const float* __restrict__ p0, const float* __restrict__ p1,
                  const float* __restrict__ p2,
                  const float4* __restrict__ boxes,
                  const float* __restrict__ gt_boxes, const int* __restrict__ gt_labels,
                  float* __restrict__ lboxOut, float* __restrict__ corrOut,
                  float* __restrict__ nposOut){
  const int b = blockIdx.x;
  const int tid = threadIdx.x;
  const float4* bx = boxes + (size_t)b*T_ANCH;

  __shared__ float gx1[NGT],gy1[NGT],gx2[NGT],gy2[NGT],gcx[NGT],gcy[NGT],garea[NGT];
  __shared__ int   glab[NGT];
  __shared__ float distS[T_ANCH];
  __shared__ int   candT[NCAND];
  __shared__ float4 candBox[NCAND];
  __shared__ float cost[NCOST];
  __shared__ float sVal[256];
  __shared__ int   sIdx[256];
  __shared__ int   assignedSlot[NGT];   // slot index (0..199) or -1
  __shared__ int   assignedT[NGT];      // anchor id assigned to gt n
  __shared__ int   chosenAnchor[NGT];   // anchors consumed so far
  __shared__ int   doneFlag;
  __shared__ float shLb[NGT], shCc[NGT], shNp[NGT];

  if (tid < NGT){
    const float* g = gt_boxes + ((size_t)b*NGT + tid)*4;
    float x1=g[0],y1=g[1],x2=g[2],y2=g[3];
    gx1[tid]=x1; gy1[tid]=y1; gx2[tid]=x2; gy2[tid]=y2;
    gcx[tid]=(x1+x2)*0.5f; gcy[tid]=(y1+y2)*0.5f;
    garea[tid]=fmaxf(x2-x1,0.f)*fmaxf(y2-y1,0.f);
    glab[tid]=gt_labels[b*NGT+tid];
    assignedSlot[tid]=-1; assignedT[tid]=-1; chosenAnchor[tid]=-1;
  }
  __syncthreads();

  // ---- top-10 nearest anchors per gt (matches lax.top_k tie-breaking) ----
  for (int n=0;n<NGT;n++){
    float cx=gcx[n], cy=gcy[n];
    for (int t=tid;t<T_ANCH;t+=256){
      float4 bb=bx[t];
      float dx=bb.x-cx, dy=bb.y-cy;
      distS[t]=dx*dx+dy*dy;
    }
    __syncthreads();
    for (int r=0;r<TOPK;r++){
      float lv=FLT_MAX; int li=0x7fffffff;
      for (int t=tid;t<T_ANCH;t+=256){
        float v=distS[t];
        if (v<lv){ lv=v; li=t; }          // ascending t keeps smallest index on ties
      }
      sVal[tid]=lv; sIdx[tid]=li;
      __syncthreads();
      for (int off=128;off>0;off>>=1){
        if (tid<off){
          float v2=sVal[tid+off]; int i2=sIdx[tid+off];
          if (v2<sVal[tid] || (v2==sVal[tid] && i2<sIdx[tid])){ sVal[tid]=v2; sIdx[tid]=i2; }
        }
        __syncthreads();
      }
      if (tid==0){ int tb=sIdx[0]; candT[n*TOPK+r]=tb; distS[tb]=FLT_MAX; }
      __syncthreads();
    }
  }

  if (tid < NCAND) candBox[tid]=bx[candT[tid]];
  __syncthreads();

  // ---- cost matrix: cost[i*NGT+n] = 1.5*cls + 6*(1-iou) ----
  for (int e=tid;e<NCOST;e+=256){
    int i=e/NGT, n=e-(e/NGT)*NGT;
    float4 cb=candBox[i];
    float cx1=cb.x-cb.z*0.5f, cy1=cb.y-cb.w*0.5f;
    float cx2=cb.x+cb.z*0.5f, cy2=cb.y+cb.w*0.5f;
    float a1=fmaxf(cx2-cx1,0.f)*fmaxf(cy2-cy1,0.f);
    float iw=fmaxf(fminf(cx2,gx2[n])-fmaxf(cx1,gx1[n]),0.f);
    float ih=fmaxf(fminf(cy2,gy2[n])-fmaxf(cy1,gy1[n]),0.f);
    float inter=iw*ih;
    float uni=a1+garea[n]-inter+EPSF;
    float iou=fminf(fmaxf(inter/uni,0.f),1.f);
    float sel=logit_at(p0,p1,p2,b,candT[i],glab[n]);
    float pr=fminf(fmaxf(sigm(sel),1e-6f),1.0f-1e-6f);
    cost[e]=1.5f*(-logf(pr)) + 6.0f*(1.0f-iou);
  }
  __syncthreads();

  // ---- greedy assignment: repeatedly take lexicographic-min valid pair ----
  for (int round=0; round<=NGT; ++round){
    float lv=FLT_MAX; int li=-1;
    for (int e=tid;e<NCOST;e+=256){
      int i=e/NGT, n=e-(e/NGT)*NGT;
      if (assignedSlot[n]>=0) continue;
      int a=candT[i];
      bool used=false;
      #pragma unroll
      for (int k=0;k<NGT;k++) used = used || (chosenAnchor[k]==a);
      if (used) continue;
      float cv=cost[e];
      if (cv<lv || (cv==lv && e<li)){ lv=cv; li=e; }
    }
    sVal[tid]=lv; sIdx[tid]=li;
    __syncthreads();
    for (int off=128;off>0;off>>=1){
      if (tid<off){
        float v2=sVal[tid+off]; int i2=sIdx[tid+off];
        bool take = (i2>=0) && (sIdx[tid]<0 || v2<sVal[tid] ||
                                (v2==sVal[tid] && i2<sIdx[tid]));
        if (take){ sVal[tid]=v2; sIdx[tid]=i2; }
      }
      __syncthreads();
    }
    if (tid==0){
      int e=sIdx[0];
      if (e<0) doneFlag=1;
      else {
        doneFlag=0;
        int i=e/NGT, n=e-(e/NGT)*NGT;
        assignedSlot[n]=i; assignedT[n]=candT[i]; chosenAnchor[round]=candT[i];
      }
    }
    __syncthreads();
    if (doneFlag) break;
  }

  // ---- per-gt box loss + focal correction ----
  if (tid < NGT){
    float l=0.f, cc=0.f, np=0.f;
    if (assignedSlot[tid]>=0){
      int t=assignedT[tid];
      float4 pb=bx[t];
      float px1=pb.x-pb.z*0.5f, py1=pb.y-pb.w*0.5f;
      float px2=pb.x+pb.z*0.5f, py2=pb.y+pb.w*0.5f;
      l = 1.0f - bbox_ciou_f(px1,py1,px2,py2, gx1[tid],gy1[tid],gx2[tid],gy2[tid]);
      float x = logit_at(p0,p1,p2,b,t,glab[tid]);
      cc = focal1f(x) - focal0f(x);
      np = 1.0f;
    }
    shLb[tid]=l; shCc[tid]=cc; shNp[tid]=np;
  }
  __syncthreads();
  if (tid==0){
    float lb=0.f, cc=0.f, np=0.f;
    for (int n=0;n<NGT;n++){ lb+=shLb[n]; cc+=shCc[n]; np+=shNp[n]; }
    lboxOut[b]=lb; corrOut[b]=cc; nposOut[b]=np;
  }
}

// =====================================================================
// Kernel C: deterministic final reduction -> scalar loss.
// =====================================================================
__global__ __launch_bounds__(256)
void finalize_kernel(const float* __restrict__ blockSums, int nBlocks,
                     const float* __restrict__ lbox, const float* __restrict__ corr,
                     const float* __restrict__ npos, float* __restrict__ out){
  __shared__ float red[256];
  int tid=threadIdx.x;
  float a=0.f;
  for (int i=tid;i<nBlocks;i+=256) a+=blockSums[i];
  red[tid]=a;
  __syncthreads();
  for (int off=128;off>0;off>>=1){
    if (tid<off) red[tid]+=red[tid+off];
    __syncthreads();
  }
  if (tid==0){
    float f0=red[0];
    float lb=0.f, cc=0.f, np=0.f;
    for (int i=0;i<BATCH;i++){ lb+=lbox[i]; cc+=corr[i]; np+=npos[i]; }
    float denom=fmaxf(np,1.0f);
    out[0]=7.5f*lb/denom + 0.5f*(f0+cc)/denom;
  }
}

// =====================================================================
extern "C" void kernel_launch(void* const* d_in, const int* in_sizes, int n_in,
                              void* d_out, int out_size, void* d_ws, size_t ws_size,
                              hipStream_t stream){
  const float* p0  = (const float*)d_in[0];
  const float* p1  = (const float*)d_in[1];
  const float* p2  = (const float*)d_in[2];
  const float* gtb = (const float*)d_in[3];
  const int*   gtl = (const int*)d_in[4];
  float* out = (float*)d_out;

  char* ws = (char*)d_ws;
  size_t off = 0;
  float4* boxes = (float4*)(ws + off); off += (size_t)BATCH*T_ANCH*sizeof(float4);
  const int blocksA = (BATCH*T_ANCH)/ABLK;           // 2100 exactly
  float* blockSums = (float*)(ws + off); off += ((size_t)blocksA*sizeof(float)+15)&~(size_t)15;
  float* lbox = (float*)(ws + off); off += 128;
  float* corr = (float*)(ws + off); off += 128;
  float* npos = (float*)(ws + off); off += 128;

  decode_focal_kernel<<<blocksA, ABLK, 0, stream>>>(p0,p1,p2,boxes,blockSums);
  match_kernel<<<BATCH, 256, 0, stream>>>(p0,p1,p2,boxes,gtb,gtl,lbox,corr,npos);
  finalize_kernel<<<1, 256, 0, stream>>>(blockSums, blocksA, lbox, corr, npos, out);
}